// LiteGearNet_31688268709972
// MI455X (gfx1250) — compile-verified
//
#include <hip/hip_runtime.h>
#include <hip/hip_bf16.h>
#include <stdint.h>

#define DIM 128
#define LN_EPS 1e-5f

typedef __attribute__((ext_vector_type(2))) float v2f;
typedef __attribute__((ext_vector_type(8))) float v8f;

// ---------------------------------------------------------------------------
// utility kernels
// ---------------------------------------------------------------------------
__global__ void zero_f32(float* __restrict__ p, int n) {
    int i = blockIdx.x * blockDim.x + threadIdx.x;
    if (i < n) p[i] = 0.0f;
}

__global__ void deg_kernel(const int64_t* __restrict__ dst, float* __restrict__ deg, int nE) {
    int e = blockIdx.x * blockDim.x + threadIdx.x;
    if (e < nE) atomicAdd(&deg[(int)dst[e]], 1.0f);
}

// one wave (32 lanes) per edge: float4 load per lane (512B coalesced), 4 atomics
__global__ void scatter_edges(const float* __restrict__ H,
                              const int64_t* __restrict__ src,
                              const int64_t* __restrict__ dst,
                              float* __restrict__ M, int nE) {
    int e = blockIdx.x * (blockDim.x >> 5) + (threadIdx.x >> 5);
    if (e >= nE) return;
    int lane = threadIdx.x & 31;
    const float* hs = H + (size_t)src[e] * DIM;
    float*       md = M + (size_t)dst[e] * DIM;
    float4 v = ((const float4*)hs)[lane];
    atomicAdd(&md[lane * 4 + 0], v.x);
    atomicAdd(&md[lane * 4 + 1], v.y);
    atomicAdd(&md[lane * 4 + 2], v.z);
    atomicAdd(&md[lane * 4 + 3], v.w);
}

// ---------------------------------------------------------------------------
// Cooperative load of W[128][128] into LDS in k-pair-interleaved layout:
//   sW[kp*256 + n*2 + (k&1)],  kp = k>>1
// so the B fragment {W[ka][n], W[ka+1][n]} is one aligned 8-byte pair
// -> single ds_load_b64 directly into the even-aligned VGPR pair WMMA wants.
// ---------------------------------------------------------------------------
__device__ __forceinline__ void load_W_interleaved(const float* __restrict__ W,
                                                   float* __restrict__ sW) {
    const float4* Wv = (const float4*)W;
    for (int i = threadIdx.x; i < DIM * DIM / 4; i += blockDim.x) {
        float4 w  = Wv[i];           // row k, cols n4..n4+3
        int    k  = i >> 5;          // i / (DIM/4)
        int    n4 = (i & 31) * 4;
        int    kp = k >> 1;
        int    kb = k & 1;
        float* b  = &sW[kp * 256 + n4 * 2 + kb];
        b[0] = w.x;  b[2] = w.y;  b[4] = w.z;  b[6] = w.w;
    }
}

// ---------------------------------------------------------------------------
// WMMA f32 GEMM: out[M,128] = op(A[M,128] (optionally *1/deg)) @ W[128,128] + bias
// One wave computes a 16-row strip across all 128 columns (8 n-tiles).
// ---------------------------------------------------------------------------
template <bool SCALE_DEG, bool RELU>
__global__ __launch_bounds__(256) void gemm128(const float* __restrict__ A,
                                               const float* __restrict__ W,
                                               const float* __restrict__ bias,
                                               const float* __restrict__ deg,
                                               float* __restrict__ out, int nrows) {
    __shared__ float sW[DIM * DIM];  // 64 KB (WGP has 320 KB LDS)
    load_W_interleaved(W, sW);
    __syncthreads();

    const int wave  = threadIdx.x >> 5;
    const int lane  = threadIdx.x & 31;
    const int half  = lane >> 4;   // 0: K pair {0,1}, 1: K pair {2,3}
    const int l16   = lane & 15;
    const int mbase = (blockIdx.x * 8 + wave) * 16;
    if (mbase >= nrows) return;  // wave-uniform

    const int row  = mbase + l16;
    const int rowc = row < nrows ? row : nrows - 1;  // clamp loads, guard stores

    float scale = 1.0f;
    if (SCALE_DEG) {
        float d = deg[rowc];
        scale   = 1.0f / (d > 1.0f ? d : 1.0f);
    }

    v8f acc[8];
#pragma unroll
    for (int t = 0; t < 8; ++t) {
        float b = bias[t * 16 + l16];
#pragma unroll
        for (int v = 0; v < 8; ++v) acc[t][v] = b;
    }

    const float* arow = A + (size_t)rowc * DIM;
    for (int k = 0; k < DIM; k += 4) {
        const int ka = k + half * 2;           // even -> 8B aligned
        v2f a = *(const v2f*)&arow[ka];        // global_load_b64
        if (SCALE_DEG) { a[0] *= scale; a[1] *= scale; }
        const int kp = ka >> 1;
#pragma unroll
        for (int t = 0; t < 8; ++t) {
            v2f b  = *(const v2f*)&sW[kp * 256 + (t * 16 + l16) * 2];  // ds_load_b64
            acc[t] = __builtin_amdgcn_wmma_f32_16x16x4_f32(
                false, a, false, b, (short)0, acc[t], false, false);
        }
    }

    // C/D layout: lane, vgpr v  ->  m = mbase + v + 8*half,  n = t*16 + l16
#pragma unroll
    for (int v = 0; v < 8; ++v) {
        int m = mbase + v + 8 * half;
        if (m >= nrows) continue;
        float* orow = out + (size_t)m * DIM;
#pragma unroll
        for (int t = 0; t < 8; ++t) {
            float x = acc[t][v];
            if (RELU) x = x > 0.0f ? x : 0.0f;
            orow[t * 16 + l16] = x;
        }
    }
}

// ---------------------------------------------------------------------------
// WMMA GEMM + residual + LayerNorm, fused:  H = LN(H + T@W + bias) * g + b
// Row m's 128 values live in the 16 lanes of one half-wave at vgpr index m%8,
// so the LN reduction is sum over 8 tiles + shfl_xor over masks {1,2,4,8}.
// ---------------------------------------------------------------------------
__global__ __launch_bounds__(256) void gemm128_res_ln(const float* __restrict__ T,
                                                      const float* __restrict__ W,
                                                      const float* __restrict__ bias,
                                                      float* __restrict__ H,
                                                      const float* __restrict__ g,
                                                      const float* __restrict__ bln,
                                                      int nrows) {
    __shared__ float sW[DIM * DIM];
    load_W_interleaved(W, sW);
    __syncthreads();

    const int wave  = threadIdx.x >> 5;
    const int lane  = threadIdx.x & 31;
    const int half  = lane >> 4;
    const int l16   = lane & 15;
    const int mbase = (blockIdx.x * 8 + wave) * 16;
    if (mbase >= nrows) return;

    const int row  = mbase + l16;
    const int rowc = row < nrows ? row : nrows - 1;

    v8f acc[8];
#pragma unroll
    for (int t = 0; t < 8; ++t) {
        float b = bias[t * 16 + l16];
#pragma unroll
        for (int v = 0; v < 8; ++v) acc[t][v] = b;
    }

    const float* arow = T + (size_t)rowc * DIM;
    for (int k = 0; k < DIM; k += 4) {
        const int ka = k + half * 2;
        v2f a = *(const v2f*)&arow[ka];
        const int kp = ka >> 1;
#pragma unroll
        for (int t = 0; t < 8; ++t) {
            v2f b  = *(const v2f*)&sW[kp * 256 + (t * 16 + l16) * 2];
            acc[t] = __builtin_amdgcn_wmma_f32_16x16x4_f32(
                false, a, false, b, (short)0, acc[t], false, false);
        }
    }

    // residual add: z = H + mlp
#pragma unroll
    for (int v = 0; v < 8; ++v) {
        int m  = mbase + v + 8 * half;
        int mc = m < nrows ? m : nrows - 1;
        const float* hrow = H + (size_t)mc * DIM;
#pragma unroll
        for (int t = 0; t < 8; ++t) acc[t][v] += hrow[t * 16 + l16];
    }

    // LayerNorm per row (v index), reduce across 8 tiles then 16 lanes
#pragma unroll
    for (int v = 0; v < 8; ++v) {
        float s = 0.0f;
#pragma unroll
        for (int t = 0; t < 8; ++t) s += acc[t][v];
        s += __shfl_xor(s, 1);
        s += __shfl_xor(s, 2);
        s += __shfl_xor(s, 4);
        s += __shfl_xor(s, 8);
        float mean = s * (1.0f / DIM);
        float q = 0.0f;
#pragma unroll
        for (int t = 0; t < 8; ++t) {
            float d = acc[t][v] - mean;
            q += d * d;
        }
        q += __shfl_xor(q, 1);
        q += __shfl_xor(q, 2);
        q += __shfl_xor(q, 4);
        q += __shfl_xor(q, 8);
        float rstd = rsqrtf(q * (1.0f / DIM) + LN_EPS);
#pragma unroll
        for (int t = 0; t < 8; ++t) acc[t][v] = (acc[t][v] - mean) * rstd;
    }

    // scale/shift + store back to H
#pragma unroll
    for (int v = 0; v < 8; ++v) {
        int m = mbase + v + 8 * half;
        if (m >= nrows) continue;
        float* hrow = H + (size_t)m * DIM;
#pragma unroll
        for (int t = 0; t < 8; ++t) {
            int n   = t * 16 + l16;
            hrow[n] = acc[t][v] * g[n] + bln[n];
        }
    }
}

// ---------------------------------------------------------------------------
// launch
// ---------------------------------------------------------------------------
extern "C" void kernel_launch(void* const* d_in, const int* in_sizes, int n_in,
                              void* d_out, int out_size, void* d_ws, size_t ws_size,
                              hipStream_t stream) {
    const float*   x    = (const float*)d_in[0];
    const int64_t* eidx = (const int64_t*)d_in[1];  // int64 per reference
    const float*   in_w = (const float*)d_in[2];
    const float*   in_b = (const float*)d_in[3];
    const float*   w1   = (const float*)d_in[4];
    const float*   b1   = (const float*)d_in[5];
    const float*   w2   = (const float*)d_in[6];
    const float*   b2   = (const float*)d_in[7];
    const float*   ln_g = (const float*)d_in[8];
    const float*   ln_b = (const float*)d_in[9];
    float*         h    = (float*)d_out;

    const int N = in_sizes[0] / DIM;
    const int E = in_sizes[1] / 2;
    const int LAYERS = in_sizes[4] / (DIM * DIM);

    const int64_t* src = eidx;
    const int64_t* dst = eidx + E;

    float* deg = (float*)d_ws;
    float* m   = deg + N;
    float* t   = m + (size_t)N * DIM;

    const int threads  = 256;
    const int gemmGrid = (N + 127) / 128;  // 8 waves * 16 rows per block

    // degree (computed once, reused every layer — same math as reference)
    zero_f32<<<(N + threads - 1) / threads, threads, 0, stream>>>(deg, N);
    deg_kernel<<<(E + threads - 1) / threads, threads, 0, stream>>>(dst, deg, E);

    // h = x @ in_w + in_b
    gemm128<false, false><<<gemmGrid, threads, 0, stream>>>(x, in_w, in_b, nullptr, h, N);

    for (int i = 0; i < LAYERS; ++i) {
        // m = segment_sum(h[src], dst)
        zero_f32<<<(N * DIM + threads - 1) / threads, threads, 0, stream>>>(m, N * DIM);
        scatter_edges<<<(E + 7) / 8, threads, 0, stream>>>(h, src, dst, m, E);
        // t = relu((m / deg) @ w1 + b1)   (1/deg folded into A-load)
        gemm128<true, true><<<gemmGrid, threads, 0, stream>>>(
            m, w1 + (size_t)i * DIM * DIM, b1 + (size_t)i * DIM, deg, t, N);
        // h = LN(h + t @ w2 + b2) * g + b
        gemm128_res_ln<<<gemmGrid, threads, 0, stream>>>(
            t, w2 + (size_t)i * DIM * DIM, b2 + (size_t)i * DIM, h,
            ln_g + (size_t)i * DIM, ln_b + (size_t)i * DIM, N);
    }
}